// Transformer_38276748542144
// MI455X (gfx1250) — compile-verified
//
#include <hip/hip_runtime.h>

typedef float v2f __attribute__((ext_vector_type(2)));
typedef float v8f __attribute__((ext_vector_type(8)));

#define ASTR 68   // row-major LDS stride (float4-aligned stores, conflict-free column reads)
#define BSTR 66   // n-major (transposed) LDS stride: even -> 8B-aligned b64 frag loads

// D = A(16x4) * B(4x16) + C  on the fp32 matrix pipe
__device__ __forceinline__ v8f wmma_f32(v2f a, v2f b, v8f c) {
    return __builtin_amdgcn_wmma_f32_16x16x4_f32(false, a, false, b, (short)0, c, false, false);
}

// ---------------------------------------------------------------------------
// x[(i*B+b)*512 + d] = z[b,t,d,h,w] + pos_emb(i,d),  i = t*64 + h*8 + w
// ---------------------------------------------------------------------------
__global__ __launch_bounds__(256) void embed_kernel(const float* __restrict__ z,
                                                    float* __restrict__ x) {
    int idx = blockIdx.x * 256 + threadIdx.x;          // over 1024*4*512 = 2M
    if (idx >= 1024 * 4 * 512) return;
    int d  = idx & 511;
    int rb = idx >> 9;
    int b  = rb & 3;
    int i  = rb >> 2;
    int t  = i >> 6;
    int hw = i & 63;
    float zv = z[(((size_t)b * 16 + t) * 512 + d) * 64 + hw];
    int j = (d < 256) ? d : d - 256;
    // inv_freq[j] = 10000^(-2j/512) = exp(-j * ln(10000)/256)
    float invf = expf(-(float)j * (9.210340371976184f / 256.0f));
    float ang  = (float)i * invf;
    float pe   = (d < 256) ? sinf(ang) : cosf(ang);
    x[idx] = zv + pe;
}

// ---------------------------------------------------------------------------
// C(MxN) = op(A(MxK) @ B(KxN) + bias), fp32 WMMA 16x16x4.
// Block = 128 threads (4 waves); block tile 128x64; wave tile 32x64.
// B stored transposed in LDS (n-major) so B-fragments are single b64 loads.
// M % 128 == 0, N % 64 == 0, K % 64 == 0.
// ---------------------------------------------------------------------------
__global__ __launch_bounds__(128) void gemm64(const float* __restrict__ A,
                                              const float* __restrict__ B,
                                              const float* __restrict__ bias,
                                              float* __restrict__ C,
                                              int M, int N, int K, int relu) {
    __shared__ float As[128 * ASTR];
    __shared__ float Bt[64 * BSTR];
    const int tid  = threadIdx.x;
    const int lane = tid & 31;
    const int wave = tid >> 5;
    const int row0 = blockIdx.y * 128;
    const int col0 = blockIdx.x * 64;
    const int m    = lane & 15;           // frag row/col within 16
    const int kk   = (lane >> 4) << 1;    // lane-half K offset (0 or 2)
    const int rofs = (lane >> 4) << 3;    // D-frag row offset (0 or 8)

    v8f acc[8];                            // [rt*4 + ct]
    for (int t = 0; t < 8; ++t)
        for (int e = 0; e < 8; ++e) acc[t][e] = 0.0f;

    for (int kc = 0; kc < K; kc += 64) {
        __syncthreads();
        // A tile: 128 rows x 64 k  (row-major, float4)
        for (int i = tid; i < 128 * 16; i += 128) {
            int r  = i >> 4;
            int c4 = (i & 15) << 2;
            float4 va = *(const float4*)(A + (size_t)(row0 + r) * K + kc + c4);
            *(float4*)&As[r * ASTR + c4] = va;
        }
        // B tile: 64 k x 64 n, stored transposed as Bt[n][k]
        for (int i = tid; i < 64 * 16; i += 128) {
            int r  = i >> 4;               // k index
            int c4 = (i & 15) << 2;        // n index
            float4 vb = *(const float4*)(B + (size_t)(kc + r) * N + col0 + c4);
            Bt[(c4 + 0) * BSTR + r] = vb.x;
            Bt[(c4 + 1) * BSTR + r] = vb.y;
            Bt[(c4 + 2) * BSTR + r] = vb.z;
            Bt[(c4 + 3) * BSTR + r] = vb.w;
        }
        __syncthreads();
        const int ar0 = (wave * 32 + m) * ASTR;
        const int ar1 = (wave * 32 + 16 + m) * ASTR;
        for (int ks = 0; ks < 64; ks += 4) {
            v2f a0, a1;
            a0.x = As[ar0 + ks + kk];
            a0.y = As[ar0 + ks + kk + 1];
            a1.x = As[ar1 + ks + kk];
            a1.y = As[ar1 + ks + kk + 1];
            for (int ct = 0; ct < 4; ++ct) {
                v2f bf;
                bf.x = Bt[(ct * 16 + m) * BSTR + ks + kk];
                bf.y = Bt[(ct * 16 + m) * BSTR + ks + kk + 1];
                acc[ct]     = wmma_f32(a0, bf, acc[ct]);
                acc[4 + ct] = wmma_f32(a1, bf, acc[4 + ct]);
            }
        }
    }
    // epilogue
    for (int rt = 0; rt < 2; ++rt)
        for (int ct = 0; ct < 4; ++ct) {
            int gcol = col0 + ct * 16 + m;
            float bv = bias ? bias[gcol] : 0.0f;
            for (int r = 0; r < 8; ++r) {
                int grow = row0 + wave * 32 + rt * 16 + rofs + r;
                float v = acc[rt * 4 + ct][r] + bv;
                if (relu) v = fmaxf(v, 0.0f);
                C[(size_t)grow * N + gcol] = v;
            }
        }
}

// ---------------------------------------------------------------------------
// Flash attention, block-causal (block = 64 = one frame). One WG per
// (q-block, b, h). q/k/v layout: [(i*4 + b)*512 + h*64 + d]. DH = 64.
// K tile kept row-major (already n-major for S = Q K^T); V tile stored
// transposed (dh-major) into the same buffer for b64 P.V fragments.
// ---------------------------------------------------------------------------
__global__ __launch_bounds__(128) void attn_kernel(const float* __restrict__ q,
                                                   const float* __restrict__ k,
                                                   const float* __restrict__ v,
                                                   float* __restrict__ av) {
    __shared__ float Qs[64 * ASTR];
    __shared__ float KVs[64 * ASTR];  // K row-major (stride ASTR), then V^T (stride BSTR)
    __shared__ float Ss[64 * ASTR];
    __shared__ float rowmax[64], rowsum[64], rowscale[64];

    const int qb   = blockIdx.x;          // 0..15
    const int b    = blockIdx.y >> 3;     // 0..3
    const int h    = blockIdx.y & 7;      // 0..7
    const int tid  = threadIdx.x;
    const int lane = tid & 31;
    const int wave = tid >> 5;
    const int m    = lane & 15;
    const int kk   = (lane >> 4) << 1;
    const int rofs = (lane >> 4) << 3;
    const float NEGBIG = -3.402823466e38f;

    v8f o[4];
    for (int t = 0; t < 4; ++t)
        for (int e = 0; e < 8; ++e) o[t][e] = 0.0f;
    if (tid < 64) { rowmax[tid] = NEGBIG; rowsum[tid] = 0.0f; }

    // load Q tile (rows qb*64 + r)
    for (int i = tid; i < 64 * 16; i += 128) {
        int r  = i >> 4;
        int c4 = (i & 15) << 2;
        float4 vq = *(const float4*)(q + ((size_t)(qb * 64 + r) * 4 + b) * 512 + h * 64 + c4);
        *(float4*)&Qs[r * ASTR + c4] = vq;
    }

    const int arow = (wave * 16 + m) * ASTR;

    for (int kb = 0; kb <= qb; ++kb) {
        __syncthreads();
        // load K tile (row-major: KVs[key][dh])
        for (int i = tid; i < 64 * 16; i += 128) {
            int r  = i >> 4;
            int c4 = (i & 15) << 2;
            float4 vk = *(const float4*)(k + ((size_t)(kb * 64 + r) * 4 + b) * 512 + h * 64 + c4);
            *(float4*)&KVs[r * ASTR + c4] = vk;
        }
        __syncthreads();
        // S = Q K^T * scale  (wave strip 16 q-rows x 64 k-cols)
        v8f s[4];
        for (int t = 0; t < 4; ++t)
            for (int e = 0; e < 8; ++e) s[t][e] = 0.0f;
        for (int ks = 0; ks < 64; ks += 4) {
            v2f a;
            a.x = Qs[arow + ks + kk];
            a.y = Qs[arow + ks + kk + 1];
            for (int ct = 0; ct < 4; ++ct) {
                v2f bf;  // B[kd][n] = K[n][kd]: adjacent in kd -> b64
                bf.x = KVs[(ct * 16 + m) * ASTR + ks + kk];
                bf.y = KVs[(ct * 16 + m) * ASTR + ks + kk + 1];
                s[ct] = wmma_f32(a, bf, s[ct]);
            }
        }
        for (int ct = 0; ct < 4; ++ct)
            for (int r = 0; r < 8; ++r)
                Ss[(wave * 16 + rofs + r) * ASTR + ct * 16 + m] = s[ct][r] * 0.125f;
        __syncthreads();
        // overwrite KVs with V^T tile: KVs[dh*BSTR + key]
        for (int i = tid; i < 64 * 16; i += 128) {
            int r  = i >> 4;               // key index
            int c4 = (i & 15) << 2;        // dh index
            float4 vv = *(const float4*)(v + ((size_t)(kb * 64 + r) * 4 + b) * 512 + h * 64 + c4);
            KVs[(c4 + 0) * BSTR + r] = vv.x;
            KVs[(c4 + 1) * BSTR + r] = vv.y;
            KVs[(c4 + 2) * BSTR + r] = vv.z;
            KVs[(c4 + 3) * BSTR + r] = vv.w;
        }
        // online softmax: one thread per query row (disjoint from KVs writes)
        if (tid < 64) {
            float* srow = &Ss[tid * ASTR];
            float tmax = srow[0];
            for (int c = 1; c < 64; ++c) tmax = fmaxf(tmax, srow[c]);
            float omax = rowmax[tid];
            float nmax = fmaxf(omax, tmax);
            float sc   = expf(omax - nmax);       // 0 on first tile
            float ps   = 0.0f;
            for (int c = 0; c < 64; ++c) {
                float e = expf(srow[c] - nmax);
                srow[c] = e;
                ps += e;
            }
            rowsum[tid]   = rowsum[tid] * sc + ps;
            rowmax[tid]   = nmax;
            rowscale[tid] = sc;
        }
        __syncthreads();
        // rescale accumulator, then O += P V
        for (int ct = 0; ct < 4; ++ct)
            for (int r = 0; r < 8; ++r)
                o[ct][r] *= rowscale[wave * 16 + rofs + r];
        for (int ks = 0; ks < 64; ks += 4) {
            v2f a;
            a.x = Ss[arow + ks + kk];
            a.y = Ss[arow + ks + kk + 1];
            for (int ct = 0; ct < 4; ++ct) {
                v2f bf;  // B[kd][n] = V[kd][n] = KVs[n*BSTR + kd]: adjacent -> b64
                bf.x = KVs[(ct * 16 + m) * BSTR + ks + kk];
                bf.y = KVs[(ct * 16 + m) * BSTR + ks + kk + 1];
                o[ct] = wmma_f32(a, bf, o[ct]);
            }
        }
    }
    __syncthreads();
    if (tid < 64) rowscale[tid] = 1.0f / rowsum[tid];
    __syncthreads();
    for (int ct = 0; ct < 4; ++ct)
        for (int r = 0; r < 8; ++r) {
            int row = wave * 16 + rofs + r;
            int i   = qb * 64 + row;
            av[((size_t)i * 4 + b) * 512 + h * 64 + ct * 16 + m] = o[ct][r] * rowscale[row];
        }
}

// ---------------------------------------------------------------------------
// x[row] += LayerNorm(y[row]) * g + b ; one wave32 per 512-col row
// ---------------------------------------------------------------------------
__global__ __launch_bounds__(256) void ln_residual(const float* __restrict__ y,
                                                   const float* __restrict__ g,
                                                   const float* __restrict__ beta,
                                                   float* __restrict__ x) {
    const int lane = threadIdx.x & 31;
    const int row  = blockIdx.x * 8 + (threadIdx.x >> 5);
    const float* yr = y + (size_t)row * 512;
    float vals[16];
    float s = 0.0f;
    for (int j = 0; j < 16; ++j) { vals[j] = yr[lane + j * 32]; s += vals[j]; }
    for (int off = 16; off >= 1; off >>= 1) s += __shfl_xor(s, off, 32);
    float mean = s * (1.0f / 512.0f);
    float vs = 0.0f;
    for (int j = 0; j < 16; ++j) { float d = vals[j] - mean; vs += d * d; }
    for (int off = 16; off >= 1; off >>= 1) vs += __shfl_xor(vs, off, 32);
    float rstd = rsqrtf(vs * (1.0f / 512.0f) + 1e-5f);
    for (int j = 0; j < 16; ++j) {
        int c = lane + j * 32;
        x[(size_t)row * 512 + c] += (vals[j] - mean) * rstd * g[c] + beta[c];
    }
}

// ---------------------------------------------------------------------------
// out[b, t, l, d, h, w] = x[(i*4+b)*512 + d],  i = t*64 + h*8 + w
// ---------------------------------------------------------------------------
__global__ __launch_bounds__(256) void writeout_kernel(const float* __restrict__ x,
                                                       float* __restrict__ out, int l) {
    int idx = blockIdx.x * 256 + threadIdx.x;
    if (idx >= 1024 * 4 * 512) return;
    int d  = idx & 511;
    int rb = idx >> 9;
    int b  = rb & 3;
    int i  = rb >> 2;
    int t  = i >> 6;
    int hw = i & 63;
    out[((((size_t)b * 16 + t) * 4 + l) * 512 + d) * 64 + hw] = x[idx];
}

extern "C" void kernel_launch(void* const* d_in, const int* in_sizes, int n_in,
                              void* d_out, int out_size, void* d_ws, size_t ws_size,
                              hipStream_t stream) {
    const float* z   = (const float*)d_in[0];
    const float* Wq  = (const float*)d_in[1];
    const float* Wk  = (const float*)d_in[2];
    const float* Wv  = (const float*)d_in[3];
    const float* Wo  = (const float*)d_in[4];
    const float* W1  = (const float*)d_in[5];
    const float* b1  = (const float*)d_in[6];
    const float* W2  = (const float*)d_in[7];
    const float* b2  = (const float*)d_in[8];
    const float* g1  = (const float*)d_in[9];
    const float* be1 = (const float*)d_in[10];
    const float* g2  = (const float*)d_in[11];
    const float* be2 = (const float*)d_in[12];
    float* out = (float*)d_out;

    float* ws = (float*)d_ws;
    const size_t SZ = 4096ull * 512;      // 2M floats / 8 MB
    float* x  = ws;                        // activations (Tp*B, 512)
    float* qb = ws + 1 * SZ;
    float* kb = ws + 2 * SZ;
    float* vb = ws + 3 * SZ;
    float* ab = ws + 4 * SZ;
    float* ff = ws + 1 * SZ;               // 4096x2048 reuses q/k/v/av region
    float* yb = ws + 5 * SZ;               // pre-LN GEMM output
    // total workspace: 6*SZ floats = 48 MB

    embed_kernel<<<8192, 256, 0, stream>>>(z, x);

    const dim3 gP(8, 32);    // N=512  GEMMs (M=4096, block tile 128x64)
    const dim3 gF(32, 32);   // N=2048 GEMM

    for (int l = 0; l < 4; ++l) {
        const float* Wq_l = Wq + (size_t)l * 512 * 512;
        const float* Wk_l = Wk + (size_t)l * 512 * 512;
        const float* Wv_l = Wv + (size_t)l * 512 * 512;
        const float* Wo_l = Wo + (size_t)l * 512 * 512;
        const float* W1_l = W1 + (size_t)l * 512 * 2048;
        const float* b1_l = b1 + (size_t)l * 2048;
        const float* W2_l = W2 + (size_t)l * 2048 * 512;
        const float* b2_l = b2 + (size_t)l * 512;

        gemm64<<<gP, 128, 0, stream>>>(x, Wq_l, nullptr, qb, 4096, 512, 512, 0);
        gemm64<<<gP, 128, 0, stream>>>(x, Wk_l, nullptr, kb, 4096, 512, 512, 0);
        gemm64<<<gP, 128, 0, stream>>>(x, Wv_l, nullptr, vb, 4096, 512, 512, 0);

        attn_kernel<<<dim3(16, 32), 128, 0, stream>>>(qb, kb, vb, ab);

        gemm64<<<gP, 128, 0, stream>>>(ab, Wo_l, nullptr, yb, 4096, 512, 512, 0);
        ln_residual<<<512, 256, 0, stream>>>(yb, g1 + l * 512, be1 + l * 512, x);

        gemm64<<<gF, 128, 0, stream>>>(x, W1_l, b1_l, ff, 4096, 2048, 512, 1);
        gemm64<<<gP, 128, 0, stream>>>(ff, W2_l, b2_l, yb, 4096, 512, 2048, 0);
        ln_residual<<<512, 256, 0, stream>>>(yb, g2 + l * 512, be2 + l * 512, x);

        writeout_kernel<<<8192, 256, 0, stream>>>(x, out, l);
    }
}